// SGC_65876208386529
// MI455X (gfx1250) — compile-verified
//
#include <hip/hip_runtime.h>

typedef __attribute__((ext_vector_type(2))) float v2f;
typedef __attribute__((ext_vector_type(8))) float v8f;

#define D        64
#define WSTRIDE  66   // padded LDS stride for transposed weights (8B aligned, low conflict)

__device__ __forceinline__ void atomAddF32(float* p, float v) {
    // hardware global_atomic_add_f32 (no CAS loop)
    unsafeAtomicAdd(p, v);
}

// ---- degree / normalization --------------------------------------------

__global__ void k_deg_init(float* __restrict__ deg, int N) {
    int i = blockIdx.x * blockDim.x + threadIdx.x;
    if (i < N) deg[i] = 1.0f;                       // self-loop weight
}

__global__ void k_deg_accum(float* __restrict__ deg, const int* __restrict__ row,
                            const float* __restrict__ w, int E) {
    int e = blockIdx.x * blockDim.x + threadIdx.x;
    if (e < E) atomAddF32(&deg[row[e]], w[e]);
}

__global__ void k_dinv(float* __restrict__ deg, int N) {
    int i = blockIdx.x * blockDim.x + threadIdx.x;
    if (i < N) {
        float d = deg[i];
        deg[i] = (d > 0.0f) ? rsqrtf(d) : 0.0f;     // now holds dinv
    }
}

// ---- one propagation hop -----------------------------------------------
// h_next[n][:] = dinv[n]^2 * h_prev[n][:]   (self-loop term == init/zero)
__global__ void k_selfloop(float* __restrict__ hn, const float* __restrict__ hp,
                           const float* __restrict__ dinv, int N) {
    int idx = blockIdx.x * blockDim.x + threadIdx.x;   // one float2 per thread
    if (idx >= N * (D / 2)) return;
    int n = idx >> 5;                                  // D/2 == 32
    float dv = dinv[n];
    float s  = dv * dv;
    float2 v = ((const float2*)hp)[idx];
    float2 o; o.x = s * v.x; o.y = s * v.y;
    ((float2*)hn)[idx] = o;
}

// h_next[row] += dinv[row]*w*dinv[col] * h_prev[col] ; 32 lanes per edge, float2/lane
__global__ void k_scatter(float* __restrict__ hn, const float* __restrict__ hp,
                          const int* __restrict__ row, const int* __restrict__ col,
                          const float* __restrict__ w, const float* __restrict__ dinv,
                          int E) {
    int t = blockIdx.x * blockDim.x + threadIdx.x;
    int e = t >> 5;
    if (e >= E) return;
    int lane = t & 31;
    int r = row[e], c = col[e];
    float wn = dinv[r] * w[e] * dinv[c];
    float2 v = ((const float2*)(hp + (size_t)c * D))[lane];
    float* dst = hn + (size_t)r * D + lane * 2;
    atomAddF32(dst + 0, wn * v.x);
    atomAddF32(dst + 1, wn * v.y);
}

// ---- dense tail: out = h @ W + bias, via V_WMMA_F32_16X16X4_F32 ---------

__global__ void __launch_bounds__(256)
k_gemm_wmma(const float* __restrict__ h, const float* __restrict__ W,
            const float* __restrict__ bias, float* __restrict__ out, int M) {
    __shared__ float Wl[D * WSTRIDE];   // transposed: Wl[n*WSTRIDE + k] = W[k][n]
    __shared__ float bl[D];

    for (int i = threadIdx.x; i < D * D; i += 256) {
        int k = i >> 6, n = i & 63;
        Wl[n * WSTRIDE + k] = W[i];
    }
    if (threadIdx.x < D) bl[threadIdx.x] = bias[threadIdx.x];
    __syncthreads();

    const int wave = threadIdx.x >> 5;
    const int lane = threadIdx.x & 31;
    const int l16  = lane & 15;
    const int kk   = (lane >> 4) << 1;                 // 0 (lanes 0-15) or 2 (lanes 16-31)
    const int m0   = (blockIdx.x * 8 + wave) * 16;
    if (m0 >= M) return;                               // wave-uniform: EXEC stays all-1s

    v8f acc[4] = {};                                   // N tiles 0..3 (16 cols each)
    const float2* arow = (const float2*)(h + (size_t)(m0 + l16) * D);

#pragma unroll
    for (int k0 = 0; k0 < D; k0 += 4) {
        // A fragment: 16x4 f32; lane -> M=l16, VGPR0/1 -> K = k0+kk, k0+kk+1
        float2 af = arow[(k0 + kk) >> 1];
        v2f a; a.x = af.x; a.y = af.y;
#pragma unroll
        for (int j = 0; j < 4; ++j) {
            // B fragment: 4x16 f32; lane -> N, VGPR0/1 -> K = k0+kk, k0+kk+1
            int n = j * 16 + l16;
            float2 bf = *(const float2*)(&Wl[n * WSTRIDE + k0 + kk]);
            v2f b; b.x = bf.x; b.y = bf.y;
            acc[j] = __builtin_amdgcn_wmma_f32_16x16x4_f32(
                /*neg_a=*/false, a, /*neg_b=*/false, b,
                /*c_mod=*/(short)0, acc[j], /*reuse_a=*/false, /*reuse_b=*/false);
        }
    }

    // C/D layout: lane -> N=l16 (+16*j), VGPR r -> M = m0 + r + (lane<16 ? 0 : 8)
    const int mbase = m0 + ((lane < 16) ? 0 : 8);
#pragma unroll
    for (int j = 0; j < 4; ++j) {
        int n = j * 16 + l16;
        float bv = bl[n];
#pragma unroll
        for (int r = 0; r < 8; ++r) {
            out[(size_t)(mbase + r) * D + n] = acc[j][r] + bv;
        }
    }
}

// ---- launch -------------------------------------------------------------

extern "C" void kernel_launch(void* const* d_in, const int* in_sizes, int n_in,
                              void* d_out, int out_size, void* d_ws, size_t ws_size,
                              hipStream_t stream) {
    const float* x    = (const float*)d_in[0];   // (N, 64)
    const int*   ei   = (const int*)  d_in[1];   // (2, E)
    const float* ew   = (const float*)d_in[2];   // (E,)
    const float* Wk   = (const float*)d_in[3];   // (64, 64)
    const float* bias = (const float*)d_in[4];   // (64,)
    float*       out  = (float*)d_out;

    const int E = in_sizes[2];
    const int N = in_sizes[0] / D;
    const int* row = ei;
    const int* col = ei + E;

    // workspace carve: dinv (N f32) | h0 (N*64 f32) | h1 (N*64 f32)
    char*  ws     = (char*)d_ws;
    float* deg    = (float*)ws;                  // deg -> dinv in place
    size_t off    = (((size_t)N * sizeof(float)) + 255) & ~(size_t)255;
    size_t hbytes = (size_t)N * D * sizeof(float);
    float* h0 = (float*)(ws + off);
    float* h1 = (float*)(ws + off + hbytes);

    const int B = 256;
    k_deg_init <<<(N + B - 1) / B, B, 0, stream>>>(deg, N);
    k_deg_accum<<<(E + B - 1) / B, B, 0, stream>>>(deg, row, ew, E);
    k_dinv     <<<(N + B - 1) / B, B, 0, stream>>>(deg, N);

    const int sl_blocks = (N * (D / 2) + B - 1) / B;
    const long long sc_threads = (long long)E * 32;
    const int sc_blocks = (int)((sc_threads + B - 1) / B);

    // hop 1: x -> h0
    k_selfloop<<<sl_blocks, B, 0, stream>>>(h0, x, deg, N);
    k_scatter <<<sc_blocks, B, 0, stream>>>(h0, x, row, col, ew, deg, E);
    // hop 2: h0 -> h1
    k_selfloop<<<sl_blocks, B, 0, stream>>>(h1, h0, deg, N);
    k_scatter <<<sc_blocks, B, 0, stream>>>(h1, h0, row, col, ew, deg, E);

    // dense tail: out = h1 @ Wk + bias
    const int mtiles  = (N + 15) / 16;           // N=100000 -> 6250 exact
    const int gblocks = (mtiles + 7) / 8;        // 8 waves (tiles) per block
    k_gemm_wmma<<<gblocks, B, 0, stream>>>(h1, Wk, bias, out, N);
}